// Denoise_module_61220463837893
// MI455X (gfx1250) — compile-verified
//
#include <hip/hip_runtime.h>
#include <math.h>

typedef __attribute__((ext_vector_type(2))) float v2f;
typedef __attribute__((ext_vector_type(8))) float v8f;

constexpr int   Bn   = 4;
constexpr int   Ln   = 2048;
constexpr int   Nn   = 64;
constexpr int   Hc   = 256;      // hidden h
constexpr int   DinC = 512;      // 2h
constexpr int   Mm   = Ln + Nn;  // 2112
constexpr int   Dd   = 768;      // 3h
constexpr int   NH   = 4;        // heads
constexpr int   HD   = 64;       // head dim
constexpr float BIGC = 1000000.0f;
constexpr float NEGF = -3.0e38f;

// ---------------- WMMA helpers (f32 16x16x4) ----------------
static __device__ __forceinline__ v8f wmma4(v2f a, v2f b, v8f c) {
  return __builtin_amdgcn_wmma_f32_16x16x4_f32(false, a, false, b, (short)0, c, false, false);
}

// C(16x16) += A(i0.., lda) @ B(.., ldb)   (A row-major MxK, B row-major KxN)
static __device__ __forceinline__ v8f tile_nn(const float* A, int lda, const float* Bm, int ldb,
                                              int i0, int j0, int K, v8f acc) {
  int lane = threadIdx.x & 31;
  int lh = lane >> 4, l16 = lane & 15;
  const float* arow = A + (size_t)(i0 + l16) * lda;
  const float* bcol = Bm + j0 + l16;
  for (int kk = 0; kk < K; kk += 4) {
    int ka = kk + lh * 2;
    v2f a; a.x = arow[ka]; a.y = arow[ka + 1];
    v2f b; b.x = bcol[(size_t)ka * ldb]; b.y = bcol[(size_t)(ka + 1) * ldb];
    acc = wmma4(a, b, acc);
  }
  return acc;
}

// C(16x16) += A(i0..) @ Bt(j0..)^T   (both row-major, K inner: dot of rows)
static __device__ __forceinline__ v8f tile_nt(const float* A, int lda, const float* Bt, int ldb,
                                              int i0, int j0, int K, v8f acc) {
  int lane = threadIdx.x & 31;
  int lh = lane >> 4, l16 = lane & 15;
  const float* arow = A + (size_t)(i0 + l16) * lda;
  const float* brow = Bt + (size_t)(j0 + l16) * ldb;
  for (int kk = 0; kk < K; kk += 4) {
    int ka = kk + lh * 2;
    v2f a; a.x = arow[ka]; a.y = arow[ka + 1];
    v2f b; b.x = brow[ka]; b.y = brow[ka + 1];
    acc = wmma4(a, b, acc);
  }
  return acc;
}

// ---------------- input staging ----------------
__global__ void copy_concat(const float* samples, const float* relation,
                            float* sp, float* out_sd, float* out_rd) {
  size_t idx = (size_t)blockIdx.x * blockDim.x + threadIdx.x;
  if (idx >= (size_t)Bn * Mm * DinC) return;
  int c = (int)(idx % DinC);
  size_t t = idx / DinC;
  int r = (int)(t % Mm);
  int b = (int)(t / Mm);
  float v = (r < Ln) ? samples[((size_t)b * Ln + r) * DinC + c]
                     : relation[((size_t)b * Nn + (r - Ln)) * DinC + c];
  sp[idx] = v;
  if (r < Ln) out_sd[((size_t)b * Ln + r) * Dd + Hc + c] = v;
  else        out_rd[((size_t)b * Nn + (r - Ln)) * Dd + Hc + c] = v;
}

__global__ void copy_flat(const float* src, float* dst, size_t n) {
  size_t idx = (size_t)blockIdx.x * blockDim.x + threadIdx.x;
  if (idx < n) dst[idx] = src[idx];
}

// ---------------- generic GEMM: C[b] = A[b] @ W + bias ----------------
__global__ void gemm_aw_bias(const float* A, const float* W, const float* bias, float* C,
                             int rows, int Kdim, int ncols) {
  int i0 = blockIdx.x * 16, j0 = blockIdx.y * 16, b = blockIdx.z;
  const float* Ab = A + (size_t)b * rows * Kdim;
  v8f acc = {};
  acc = tile_nn(Ab, Kdim, W, ncols, i0, j0, Kdim, acc);
  int lane = threadIdx.x; int lh = lane >> 4, l16 = lane & 15;
#pragma unroll
  for (int g = 0; g < 8; ++g) {
    int i = i0 + g + 8 * lh, j = j0 + l16;
    C[((size_t)b * rows + i) * ncols + j] = acc[g] + bias[j];
  }
}

// ---------------- flash attention (per b, head, 16-row q tile) ----------------
__global__ void attn_flash(const float* q, const float* k, const float* v, float* ctx) {
  int mt = blockIdx.x, h = blockIdx.y, b = blockIdx.z;
  int lane = threadIdx.x;
  int lh = lane >> 4, l16 = lane & 15;
  const float* Q  = q + (size_t)b * Mm * Hc + h * HD;
  const float* Km = k + (size_t)b * Mm * Hc + h * HD;
  const float* V  = v + (size_t)b * Mm * Hc + h * HD;
  __shared__ float Pls[16 * 17];
  float mrun[8], lrun[8];
#pragma unroll
  for (int g = 0; g < 8; ++g) { mrun[g] = NEGF; lrun[g] = 0.f; }
  v8f O[4];
#pragma unroll
  for (int t = 0; t < 4; ++t) O[t] = (v8f){};
  int i0 = mt * 16;
  for (int jt = 0; jt < Mm / 16; ++jt) {
    int j0 = jt * 16;
    v8f S = {};
    S = tile_nt(Q, Hc, Km, Hc, i0, j0, HD, S);
    float scale[8];
#pragma unroll
    for (int g = 0; g < 8; ++g) {
      float s = S[g];
      float m = s;
      for (int off = 1; off < 16; off <<= 1) m = fmaxf(m, __shfl_xor(m, off, 32));
      float mnew = fmaxf(mrun[g], m);
      float sc = expf(mrun[g] - mnew);
      float p = expf(s - mnew);
      float ps = p;
      for (int off = 1; off < 16; off <<= 1) ps += __shfl_xor(ps, off, 32);
      lrun[g] = lrun[g] * sc + ps;
      mrun[g] = mnew;
      scale[g] = sc;
      Pls[(g + 8 * lh) * 17 + l16] = p;
    }
#pragma unroll
    for (int t = 0; t < 4; ++t)
#pragma unroll
      for (int g = 0; g < 8; ++g) O[t][g] *= scale[g];
    __syncthreads();
    const float* arow = &Pls[l16 * 17];
#pragma unroll
    for (int t = 0; t < 4; ++t) {
      v8f accO = O[t];
      for (int kk = 0; kk < 16; kk += 4) {
        int ka = kk + lh * 2;
        v2f a; a.x = arow[ka]; a.y = arow[ka + 1];
        v2f bb;
        bb.x = V[(size_t)(j0 + ka) * Hc + t * 16 + l16];
        bb.y = V[(size_t)(j0 + ka + 1) * Hc + t * 16 + l16];
        accO = wmma4(a, bb, accO);
      }
      O[t] = accO;
    }
    __syncthreads();
  }
#pragma unroll
  for (int t = 0; t < 4; ++t)
#pragma unroll
    for (int g = 0; g < 8; ++g) {
      int m = i0 + g + 8 * lh;
      int n = t * 16 + l16;
      ctx[((size_t)b * Mm + m) * Hc + h * HD + n] = O[t][g] / lrun[g];
    }
}

// ---------------- out projection, split into samples_d / relation_d ----------------
__global__ void gemm_outproj(const float* ctx, const float* Wo, const float* bo,
                             float* out_sd, float* out_rd) {
  int i0 = blockIdx.x * 16, j0 = blockIdx.y * 16, b = blockIdx.z;
  const float* Ab = ctx + (size_t)b * Mm * Hc;
  v8f acc = {};
  acc = tile_nn(Ab, Hc, Wo, Hc, i0, j0, Hc, acc);
  int lane = threadIdx.x; int lh = lane >> 4, l16 = lane & 15;
#pragma unroll
  for (int g = 0; g < 8; ++g) {
    int i = i0 + g + 8 * lh, j = j0 + l16;
    float val = acc[g] + bo[j];
    if (i < Ln) out_sd[((size_t)b * Ln + i) * Dd + j] = val;
    else        out_rd[((size_t)b * Nn + (i - Ln)) * Dd + j] = val;
  }
}

// ---------------- row squared norms ----------------
__global__ void rownorm(const float* X, float* x2, int rows, int D) {
  int r = blockIdx.x, b = blockIdx.y;
  const float* row = X + ((size_t)b * rows + r) * D;
  float s = 0.f;
  for (int c = threadIdx.x; c < D; c += 32) { float v = row[c]; s += v * v; }
  for (int off = 16; off; off >>= 1) s += __shfl_xor(s, off, 32);
  if (threadIdx.x == 0) x2[(size_t)b * rows + r] = s;
}

// ---------------- adjacency graph ----------------
__global__ void graph_kernel(const float* sd, const float* x2, float* graph) {
  int i0 = blockIdx.x * 16, j0 = blockIdx.y * 16, b = blockIdx.z;
  const float* X = sd + (size_t)b * Ln * Dd;
  v8f acc = {};
  acc = tile_nt(X, Dd, X, Dd, i0, j0, Dd, acc);
  int lane = threadIdx.x; int lh = lane >> 4, l16 = lane & 15;
#pragma unroll
  for (int g = 0; g < 8; ++g) {
    int i = i0 + g + 8 * lh, j = j0 + l16;
    float d2 = fmaxf(x2[(size_t)b * Ln + i] - 2.f * acc[g] + x2[(size_t)b * Ln + j], 0.f);
    float gv = -sqrtf(d2 + 1e-6f);
    if (i == j) gv -= BIGC;
    graph[((size_t)b * Ln + i) * Ln + j] = gv;
  }
}

// ---------------- kmean: masked row-max (sample_w) ----------------
__global__ void samplew_kernel(const float* label, const float* graph, float* sw) {
  int i0 = blockIdx.x * 16, b = blockIdx.y;
  const float* Lb = label + (size_t)b * Ln * Nn;
  const float* Gb = graph + (size_t)b * Ln * Ln;
  int lane = threadIdx.x; int lh = lane >> 4, l16 = lane & 15;
  float best[8];
#pragma unroll
  for (int g = 0; g < 8; ++g) best[g] = NEGF;
  for (int jt = 0; jt < Ln / 16; ++jt) {
    v8f mk = {};
    mk = tile_nt(Lb, Nn, Lb, Nn, i0, jt * 16, Nn, mk);
#pragma unroll
    for (int g = 0; g < 8; ++g) {
      int i = i0 + g + 8 * lh, j = jt * 16 + l16;
      float m = mk[g];
      float gv = Gb[(size_t)i * Ln + j];
      float val = gv * m + (1.f - m) * (-BIGC);
      best[g] = fmaxf(best[g], val);
    }
  }
#pragma unroll
  for (int g = 0; g < 8; ++g) {
    float v = best[g];
    for (int off = 1; off < 16; off <<= 1) v = fmaxf(v, __shfl_xor(v, off, 32));
    if (l16 == 0) sw[(size_t)b * Ln + i0 + g + 8 * lh] = v;
  }
}

// ---------------- kmean: lt softmax ----------------
__global__ void lt1_kernel(const float* label, const float* sw, float* lt1) {
  int n = blockIdx.x, b = blockIdx.y;
  const float* Lb = label + (size_t)b * Ln * Nn;
  const float* swb = sw + (size_t)b * Ln;
  float mx = NEGF;
  for (int l = threadIdx.x; l < Ln; l += 32) {
    float t = Lb[(size_t)l * Nn + n];
    float v = t * swb[l] + (1.f - t) * (-BIGC);
    mx = fmaxf(mx, v);
  }
  for (int off = 16; off; off >>= 1) mx = fmaxf(mx, __shfl_xor(mx, off, 32));
  float se = 0.f;
  for (int l = threadIdx.x; l < Ln; l += 32) {
    float t = Lb[(size_t)l * Nn + n];
    float v = t * swb[l] + (1.f - t) * (-BIGC);
    se += expf(v - mx);
  }
  for (int off = 16; off; off >>= 1) se += __shfl_xor(se, off, 32);
  float inv = 1.f / se;
  for (int l = threadIdx.x; l < Ln; l += 32) {
    float t = Lb[(size_t)l * Nn + n];
    float v = t * swb[l] + (1.f - t) * (-BIGC);
    lt1[((size_t)b * Nn + n) * Ln + l] = expf(v - mx) * inv;
  }
}

// ---------------- kmean: Center = c1*lt1@sd + c2*rd ----------------
__global__ void center_kernel(const float* lt1, const float* sd, const float* rd, float* Center) {
  int i0 = blockIdx.x * 16, j0 = blockIdx.y * 16, b = blockIdx.z;
  v8f acc = {};
  acc = tile_nn(lt1 + (size_t)b * Nn * Ln, Ln, sd + (size_t)b * Ln * Dd, Dd, i0, j0, Ln, acc);
  int lane = threadIdx.x; int lh = lane >> 4, l16 = lane & 15;
  const float c1 = (Nn + 1.0f) / (Nn + 2.0f);
  const float c2 = 1.0f / (Nn + 2.0f);
#pragma unroll
  for (int g = 0; g < 8; ++g) {
    int i = i0 + g + 8 * lh, j = j0 + l16;
    Center[((size_t)b * Nn + i) * Dd + j] = c1 * acc[g] + c2 * rd[((size_t)b * Nn + i) * Dd + j];
  }
}

// ---------------- kmean: Predict = 2xy - x2 - c2 ----------------
__global__ void predict_kernel(const float* sd, const float* Center, const float* x2, const float* cnorm,
                               float* out_pred, float* out_logit, float* Pt, int iter) {
  int i0 = blockIdx.x * 16, j0 = blockIdx.y * 16, b = blockIdx.z;
  v8f acc = {};
  acc = tile_nt(sd + (size_t)b * Ln * Dd, Dd, Center + (size_t)b * Nn * Dd, Dd, i0, j0, Dd, acc);
  int lane = threadIdx.x; int lh = lane >> 4, l16 = lane & 15;
#pragma unroll
  for (int g = 0; g < 8; ++g) {
    int i = i0 + g + 8 * lh, j = j0 + l16;
    float p = 2.f * acc[g] - x2[(size_t)b * Ln + i] - cnorm[(size_t)b * Nn + j];
    size_t lo = ((size_t)b * Ln + i) * Nn + j;
    out_pred[lo] = p;
    Pt[((size_t)b * Nn + j) * Ln + i] = p;
    float contrib = 0.25f * p;
    out_logit[lo] = (iter == 0) ? contrib : out_logit[lo] + contrib;
  }
}

// ---------------- _to_one_hot while-loop (single workgroup, exact) ----------------
static __device__ __forceinline__ unsigned f2key(float f) {
  unsigned u = __float_as_uint(f);
  return (u & 0x80000000u) ? ~u : (u | 0x80000000u);
}

static __device__ void mask2_row(const float* row, float* mrow, int lane) {
  // kth(=65) largest via binary search on monotonic key space; mask = (val >= kth)
  unsigned lo = 0u, hi = 0xFFFFFFFFu;
  while (lo < hi) {
    unsigned mid = (unsigned)(((unsigned long long)lo + (unsigned long long)hi + 1ull) >> 1);
    int c = 0;
    for (int i = lane; i < Ln; i += 32) c += (f2key(row[i]) >= mid) ? 1 : 0;
    for (int off = 16; off; off >>= 1) c += __shfl_xor(c, off, 32);
    if (c >= Nn + 1) lo = mid; else hi = mid - 1;
  }
  for (int i = lane; i < Ln; i += 32) mrow[i] = (f2key(row[i]) >= lo) ? 1.f : 0.f;
}

__global__ void onehot_kernel(float* Pt, float* M2, float* label) {
  __shared__ int s_flag;
  const int tid = threadIdx.x;
  const int NT = 1024, NW = 32;
  int wave = tid >> 5, lane = tid & 31;
  for (int row = wave; row < Bn * Nn; row += NW)
    mask2_row(Pt + (size_t)row * Ln, M2 + (size_t)row * Ln, lane);
  __syncthreads();
  for (int j = 0; j < 11; ++j) {
    if (tid == 0) s_flag = 0;
    __syncthreads();
    int myflag = 0;
    for (int col = tid; col < Bn * Ln; col += NT) {
      int b = col / Ln, l = col % Ln;
      const float* M2b = M2 + (size_t)b * Nn * Ln + l;
      float s = 0.f;
      for (int n = 0; n < Nn; ++n) s += M2b[(size_t)n * Ln];
      if (s >= 2.0f) myflag = 1;
    }
    if (myflag) atomicOr(&s_flag, 1);
    __syncthreads();
    int go = s_flag;
    __syncthreads();
    if (!go) break;
    // body: column-wise conflict resolution
    for (int col = tid; col < Bn * Ln; col += NT) {
      int b = col / Ln, l = col % Ln;
      float* Ptb = Pt + (size_t)b * Nn * Ln + l;
      float* M2b = M2 + (size_t)b * Nn * Ln + l;
      float cmax = NEGF;
      for (int n = 0; n < Nn; ++n) {
        float m2 = M2b[(size_t)n * Ln];
        float p1 = Ptb[(size_t)n * Ln] - (1.f - m2) * BIGC;
        cmax = fmaxf(cmax, p1);
      }
      float s4 = 0.f;
      for (int n = 0; n < Nn; ++n) {
        float m2 = M2b[(size_t)n * Ln];
        float pt = Ptb[(size_t)n * Ln];
        float p1 = pt - (1.f - m2) * BIGC;
        s4 += ((p1 >= cmax) ? 1.f : 0.f) * m2;
      }
      float m5 = (s4 > 0.5f) ? 1.f : 0.f;
      float m6 = (s4 > 1.5f) ? 1.f : 0.f;
      for (int n = 0; n < Nn; ++n) {
        float m2 = M2b[(size_t)n * Ln];
        float pt = Ptb[(size_t)n * Ln];
        float p1 = pt - (1.f - m2) * BIGC;
        float m4 = ((p1 >= cmax) ? 1.f : 0.f) * m2 * (1.f - m6);
        float pt2 = pt - m5 * BIGC;
        Ptb[(size_t)n * Ln] = m4 * pt + (1.f - m4) * pt2;
      }
    }
    __syncthreads();
    if (j < 10) {
      for (int row = wave; row < Bn * Nn; row += NW)
        mask2_row(Pt + (size_t)row * Ln, M2 + (size_t)row * Ln, lane);
      __syncthreads();
    }
  }
  __syncthreads();
  for (size_t idx = tid; idx < (size_t)Bn * Ln * Nn; idx += NT) {
    size_t b = idx / ((size_t)Ln * Nn);
    size_t l = (idx / Nn) % Ln;
    size_t n = idx % Nn;
    label[idx] = M2[(b * Nn + n) * Ln + l];
  }
}

// ---------------- poisson: scores = s@s^T + diag(-1e5) ----------------
__global__ void gscore_kernel(const float* s, float* G) {
  int i0 = blockIdx.x * 16, j0 = blockIdx.y * 16, b = blockIdx.z;
  const float* sb = s + (size_t)b * Ln * Hc;
  v8f acc = {};
  acc = tile_nt(sb, Hc, sb, Hc, i0, j0, Hc, acc);
  int lane = threadIdx.x; int lh = lane >> 4, l16 = lane & 15;
#pragma unroll
  for (int g = 0; g < 8; ++g) {
    int i = i0 + g + 8 * lh, j = j0 + l16;
    float v = acc[g];
    if (i == j) v -= 100000.0f;
    G[((size_t)b * Ln + i) * Ln + j] = v;
  }
}

__global__ void gsoftmax_kernel(float* G, float* Dinv) {
  int i = blockIdx.x, b = blockIdx.y;
  float* row = G + ((size_t)b * Ln + i) * Ln;
  int lane = threadIdx.x;
  float mx = NEGF;
  for (int j = lane; j < Ln; j += 32) mx = fmaxf(mx, row[j]);
  for (int off = 16; off; off >>= 1) mx = fmaxf(mx, __shfl_xor(mx, off, 32));
  float se = 0.f;
  for (int j = lane; j < Ln; j += 32) se += expf(row[j] - mx);
  for (int off = 16; off; off >>= 1) se += __shfl_xor(se, off, 32);
  float inv = 1.f / se;
  float rs = 0.f;
  for (int j = lane; j < Ln; j += 32) { float p = expf(row[j] - mx) * inv; row[j] = p; rs += p; }
  for (int off = 16; off; off >>= 1) rs += __shfl_xor(rs, off, 32);
  if (lane == 0) Dinv[(size_t)b * Ln + i] = 1.f / (rs + 1e-10f);
}

__global__ void buildP(const float* G, const float* Dinv, float* P) {
  __shared__ float t[16][17];
  int j0 = blockIdx.x * 16, i0 = blockIdx.y * 16, b = blockIdx.z;
  int tx = threadIdx.x, ty = threadIdx.y;
  t[ty][tx] = G[((size_t)b * Ln + j0 + ty) * Ln + i0 + tx];
  __syncthreads();
  P[((size_t)b * Ln + i0 + ty) * Ln + j0 + tx] = Dinv[(size_t)b * Ln + i0 + ty] * t[tx][ty];
}

__global__ void labeled_kernel(const float* ps, float* labeled) {
  int idx = blockIdx.x * blockDim.x + threadIdx.x;
  if (idx >= Bn * Ln) return;
  const float* row = ps + (size_t)idx * Nn;
  float s = 0.f;
  for (int n = 0; n < Nn; ++n) s += row[n];
  labeled[idx] = (s > 0.5f) ? 1.f : 0.f;
}

__global__ void total_kernel(const float* labeled, float* TotalB) {
  int b = blockIdx.x;
  float s = 0.f;
  for (int l = threadIdx.x; l < Ln; l += 32) s += labeled[(size_t)b * Ln + l];
  for (int off = 16; off; off >>= 1) s += __shfl_xor(s, off, 32);
  if (threadIdx.x == 0) TotalB[b] = s;
}

__global__ void colsum_kernel(const float* ps, float* colsum) {
  int n = blockIdx.x, b = blockIdx.y;
  float s = 0.f;
  for (int l = threadIdx.x; l < Ln; l += 32) s += ps[((size_t)b * Ln + l) * Nn + n];
  for (int off = 16; off; off >>= 1) s += __shfl_xor(s, off, 32);
  if (threadIdx.x == 0) colsum[(size_t)b * Nn + n] = s;
}

__global__ void db_kernel(const float* ps, const float* Dinv, const float* colsum,
                          const float* TotalB, const float* labeled, float* Db) {
  size_t idx = (size_t)blockIdx.x * blockDim.x + threadIdx.x;
  if (idx >= (size_t)Bn * Ln * Nn) return;
  int n = (int)(idx % Nn);
  size_t t = idx / Nn;
  int l = (int)(t % Ln);
  int b = (int)(t / Ln);
  float avg = colsum[(size_t)b * Nn + n] / TotalB[b];
  float bm = (ps[idx] - avg) * labeled[(size_t)b * Ln + l];
  Db[idx] = Dinv[(size_t)b * Ln + l] * bm;
}

// ---------------- poisson iterate: dst = P @ src + Db ----------------
__global__ void ut_kernel(const float* P, const float* src, const float* Db, float* dst) {
  int i0 = blockIdx.x * 16, j0 = blockIdx.y * 16, b = blockIdx.z;
  v8f acc = {};
  acc = tile_nn(P + (size_t)b * Ln * Ln, Ln, src + (size_t)b * Ln * Nn, Nn, i0, j0, Ln, acc);
  int lane = threadIdx.x; int lh = lane >> 4, l16 = lane & 15;
#pragma unroll
  for (int g = 0; g < 8; ++g) {
    int i = i0 + g + 8 * lh, j = j0 + l16;
    size_t o = ((size_t)b * Ln + i) * Nn + j;
    dst[o] = acc[g] + Db[o];
  }
}

// ================= host =================
extern "C" void kernel_launch(void* const* d_in, const int* in_sizes, int n_in,
                              void* d_out, int out_size, void* d_ws, size_t ws_size,
                              hipStream_t stream) {
  const float* samples  = (const float*)d_in[0];
  const float* relation = (const float*)d_in[1];
  const float* label_in = (const float*)d_in[2];
  const float* Wq = (const float*)d_in[3];  const float* bq = (const float*)d_in[4];
  const float* Wk = (const float*)d_in[5];  const float* bk = (const float*)d_in[6];
  const float* Wv = (const float*)d_in[7];  const float* bv = (const float*)d_in[8];
  const float* Wo = (const float*)d_in[9];  const float* bo = (const float*)d_in[10];
  const float* Wp = (const float*)d_in[11]; const float* bp = (const float*)d_in[12];
  (void)in_sizes; (void)n_in; (void)out_size; (void)ws_size;

  float* out = (float*)d_out;
  float* out_pred  = out;
  float* out_logit = out_pred + (size_t)Bn * Ln * Nn;
  float* out_sd    = out_logit + (size_t)Bn * Ln * Nn;
  float* out_rd    = out_sd + (size_t)Bn * Ln * Dd;
  float* out_lab1  = out_rd + (size_t)Bn * Nn * Dd;

  float* W = (float*)d_ws;
  size_t o = 0;
  auto alloc = [&](size_t n) { float* p = W + o; o += n; return p; };
  float* sp     = alloc((size_t)Bn * Mm * DinC);
  float* qb     = alloc((size_t)Bn * Mm * Hc);
  float* kb     = alloc((size_t)Bn * Mm * Hc);
  float* vb     = alloc((size_t)Bn * Mm * Hc);
  float* ctx    = alloc((size_t)Bn * Mm * Hc);
  float* graph  = alloc((size_t)Bn * Ln * Ln);   // reused as G
  float* Pbuf   = alloc((size_t)Bn * Ln * Ln);
  float* x2     = alloc((size_t)Bn * Ln);
  float* sw     = alloc((size_t)Bn * Ln);
  float* lt1    = alloc((size_t)Bn * Nn * Ln);
  float* Center = alloc((size_t)Bn * Nn * Dd);
  float* cnorm  = alloc((size_t)Bn * Nn);
  float* labelw = alloc((size_t)Bn * Ln * Nn);
  float* Pt     = alloc((size_t)Bn * Nn * Ln);
  float* M2     = alloc((size_t)Bn * Nn * Ln);
  float* sbuf   = alloc((size_t)Bn * Ln * Hc);
  float* Dinv   = alloc((size_t)Bn * Ln);
  float* colsum = alloc((size_t)Bn * Nn);
  float* TotalB = alloc((size_t)Bn);
  float* labeled= alloc((size_t)Bn * Ln);
  float* Dbb    = alloc((size_t)Bn * Ln * Nn);
  float* utA    = alloc((size_t)Bn * Ln * Nn);
  float* utB    = alloc((size_t)Bn * Ln * Nn);

  // stage inputs
  {
    size_t total = (size_t)Bn * Mm * DinC;
    copy_concat<<<dim3((unsigned)((total + 255) / 256)), 256, 0, stream>>>(samples, relation, sp, out_sd, out_rd);
  }
  {
    size_t total = (size_t)Bn * Ln * Nn;
    copy_flat<<<dim3((unsigned)((total + 255) / 256)), 256, 0, stream>>>(label_in, labelw, total);
  }
  // GAT
  dim3 gq(Mm / 16, Hc / 16, Bn);
  gemm_aw_bias<<<gq, 32, 0, stream>>>(sp, Wq, bq, qb, Mm, DinC, Hc);
  gemm_aw_bias<<<gq, 32, 0, stream>>>(sp, Wk, bk, kb, Mm, DinC, Hc);
  gemm_aw_bias<<<gq, 32, 0, stream>>>(sp, Wv, bv, vb, Mm, DinC, Hc);
  attn_flash<<<dim3(Mm / 16, NH, Bn), 32, 0, stream>>>(qb, kb, vb, ctx);
  gemm_outproj<<<dim3(Mm / 16, Hc / 16, Bn), 32, 0, stream>>>(ctx, Wo, bo, out_sd, out_rd);
  // adjacency
  rownorm<<<dim3(Ln, Bn), 32, 0, stream>>>(out_sd, x2, Ln, Dd);
  graph_kernel<<<dim3(Ln / 16, Ln / 16, Bn), 32, 0, stream>>>(out_sd, x2, graph);
  // kmeans (4 iterations)
  for (int it = 0; it < 4; ++it) {
    samplew_kernel<<<dim3(Ln / 16, Bn), 32, 0, stream>>>(labelw, graph, sw);
    lt1_kernel<<<dim3(Nn, Bn), 32, 0, stream>>>(labelw, sw, lt1);
    center_kernel<<<dim3(Nn / 16, Dd / 16, Bn), 32, 0, stream>>>(lt1, out_sd, out_rd, Center);
    rownorm<<<dim3(Nn, Bn), 32, 0, stream>>>(Center, cnorm, Nn, Dd);
    predict_kernel<<<dim3(Ln / 16, Nn / 16, Bn), 32, 0, stream>>>(out_sd, Center, x2, cnorm,
                                                                  out_pred, out_logit, Pt, it);
    onehot_kernel<<<1, 1024, 0, stream>>>(Pt, M2, labelw);
  }
  // poisson
  gemm_aw_bias<<<dim3(Ln / 16, Hc / 16, Bn), 32, 0, stream>>>(out_sd, Wp, bp, sbuf, Ln, Dd, Hc);
  gscore_kernel<<<dim3(Ln / 16, Ln / 16, Bn), 32, 0, stream>>>(sbuf, graph);
  gsoftmax_kernel<<<dim3(Ln, Bn), 32, 0, stream>>>(graph, Dinv);
  buildP<<<dim3(Ln / 16, Ln / 16, Bn), dim3(16, 16), 0, stream>>>(graph, Dinv, Pbuf);
  labeled_kernel<<<dim3((Bn * Ln + 255) / 256), 256, 0, stream>>>(labelw, labeled);
  total_kernel<<<dim3(Bn), 32, 0, stream>>>(labeled, TotalB);
  colsum_kernel<<<dim3(Nn, Bn), 32, 0, stream>>>(labelw, colsum);
  {
    size_t total = (size_t)Bn * Ln * Nn;
    db_kernel<<<dim3((unsigned)((total + 255) / 256)), 256, 0, stream>>>(labelw, Dinv, colsum, TotalB, labeled, Dbb);
  }
  // ut0 = Db counts as iteration 1; five more GEMM iterations (T=6)
  dim3 gu(Ln / 16, Nn / 16, Bn);
  ut_kernel<<<gu, 32, 0, stream>>>(Pbuf, Dbb, Dbb, utA);
  ut_kernel<<<gu, 32, 0, stream>>>(Pbuf, utA, Dbb, utB);
  ut_kernel<<<gu, 32, 0, stream>>>(Pbuf, utB, Dbb, utA);
  ut_kernel<<<gu, 32, 0, stream>>>(Pbuf, utA, Dbb, utB);
  ut_kernel<<<gu, 32, 0, stream>>>(Pbuf, utB, Dbb, out_lab1);
}